// TopKUnpool_3504693314189
// MI455X (gfx1250) — compile-verified
//
#include <hip/hip_runtime.h>
#include <cstdint>

// Problem dims (fixed by the reference): B=8, N=4096, K=2048, C=256.
#define B_    8
#define N_    4096
#define K_    2048
#define C_    256
#define C4_   (C_ / 4)            // 64 float4 per row
#define SCAN_T 256
#define PER_   (N_ / SCAN_T)      // 16 idx entries per scan thread
#define ROWS_PER_BLK 4
#define SCAT_T (ROWS_PER_BLK * C4_)  // 256 threads

#define AS1 __attribute__((address_space(1)))
#define AS3 __attribute__((address_space(3)))

// 128-bit payload type matching the async-LDS builtins' signatures:
// (v4i addrspace(1)*, v4i addrspace(3)*, imm offset, imm cpol)
typedef int v4i __attribute__((ext_vector_type(4)));
typedef AS1 v4i* g128_t;  // pointer to 128b in global memory
typedef AS3 v4i* l128_t;  // pointer to 128b in LDS

#if __has_builtin(__builtin_amdgcn_global_load_async_to_lds_b128) && \
    __has_builtin(__builtin_amdgcn_global_store_async_from_lds_b128)
#define HAVE_ASYNC_LDS 1
#else
#define HAVE_ASYNC_LDS 0
#endif

static __device__ __forceinline__ void wait_asynccnt0() {
#if __has_builtin(__builtin_amdgcn_s_wait_asynccnt)
    __builtin_amdgcn_s_wait_asynccnt(0);
#else
    asm volatile("s_wait_asynccnt 0" ::: "memory");
#endif
}

// ---------------------------------------------------------------------------
// Kernel 1: per-batch exclusive prefix count of idx>0.
// rank[b*N + n] = (#set positions before n) if idx[b,n]>0 and < K, else -1.
// One block per batch, 256 threads, 16 entries each, LDS tree scan.
// ---------------------------------------------------------------------------
__global__ void __launch_bounds__(SCAN_T)
topk_rank_kernel(const int* __restrict__ idx, int* __restrict__ rank) {
    const int b = blockIdx.x;
    const int t = threadIdx.x;
    const int* row  = idx  + (size_t)b * N_;
    int*       rrow = rank + (size_t)b * N_;

    const int base = t * PER_;
    int local[PER_];
    int cnt = 0;
#pragma unroll
    for (int i = 0; i < PER_; ++i) {
        int v = row[base + i];
        local[i] = v;
        cnt += (v > 0) ? 1 : 0;
    }

    __shared__ int part[SCAN_T];
    part[t] = cnt;
    __syncthreads();
    // Inclusive tree scan over the 256 per-thread counts.
    for (int off = 1; off < SCAN_T; off <<= 1) {
        int v = 0;
        if (t >= off) v = part[t - off];
        __syncthreads();
        if (t >= off) part[t] += v;
        __syncthreads();
    }
    int prefix = (t == 0) ? 0 : part[t - 1];

#pragma unroll
    for (int i = 0; i < PER_; ++i) {
        if (local[i] > 0) {
            rrow[base + i] = (prefix < K_) ? prefix : -1;
            ++prefix;
        } else {
            rrow[base + i] = -1;
        }
    }
}

// ---------------------------------------------------------------------------
// Kernel 2: row scatter. 4 output rows per block; 64 lanes per row, each lane
// moves one float4 (16B) => 1KB per row, fully coalesced B128 traffic.
// Valid rows stream through LDS via CDNA5 async global<->LDS ops (ASYNCcnt);
// each lane owns a private LDS slot, and waves are uniform per row, so a
// per-wave s_wait_asynccnt 0 between load and store is the only sync needed.
// Invalid rows are zero-filled with direct B128 stores.
// ---------------------------------------------------------------------------
__global__ void __launch_bounds__(SCAT_T)
topk_scatter_kernel(const float* __restrict__ x, const int* __restrict__ rank,
                    float* __restrict__ out) {
    const int t    = threadIdx.x;
    const int r    = t >> 6;        // row within block (0..3)
    const int lane = t & (C4_ - 1); // float4 lane within row (0..63)

    const long grow = (long)blockIdx.x * ROWS_PER_BLK + r;  // global row in [0, B*N)
    const int  b    = (int)(grow / N_);
    const int  sel  = rank[grow];

    float4* dst = (float4*)out + grow * (long)C4_ + lane;

#if HAVE_ASYNC_LDS
    __shared__ float4 tile[SCAT_T];
    if (sel >= 0) {
        const float4* src = (const float4*)x + ((long)b * K_ + sel) * C4_ + lane;
        // LDS offset = low 32 bits of the generic shared-memory address.
        uint32_t lds_off = (uint32_t)(uintptr_t)&tile[t];
        l128_t lp = (l128_t)(uintptr_t)lds_off;
        __builtin_amdgcn_global_load_async_to_lds_b128(
            (g128_t)(uintptr_t)src, lp, 0, 0);
        wait_asynccnt0();  // per-wave: this lane's LDS slot is now valid
        __builtin_amdgcn_global_store_async_from_lds_b128(
            (g128_t)(uintptr_t)dst, lp, 0, 0);
        // S_ENDPGM performs an implicit wait-idle; stores drain before exit.
    } else {
        *dst = make_float4(0.f, 0.f, 0.f, 0.f);
    }
#else
    if (sel >= 0) {
        const float4* src = (const float4*)x + ((long)b * K_ + sel) * C4_ + lane;
        *dst = *src;
    } else {
        *dst = make_float4(0.f, 0.f, 0.f, 0.f);
    }
#endif
}

// ---------------------------------------------------------------------------
// Launch: inputs are [x (B,K,C) f32, idx (B,N) i32, A (B,N,N) f32 -- unused].
// Output: (B,N,C) f32. d_ws holds the B*N int rank array (128 KB).
// ---------------------------------------------------------------------------
extern "C" void kernel_launch(void* const* d_in, const int* in_sizes, int n_in,
                              void* d_out, int out_size, void* d_ws, size_t ws_size,
                              hipStream_t stream) {
    (void)in_sizes; (void)n_in; (void)out_size; (void)ws_size;
    const float* x   = (const float*)d_in[0];
    const int*   idx = (const int*)d_in[1];
    // d_in[2] (A) is shape-only in the reference; never dereferenced.
    float* out = (float*)d_out;
    int*   rank = (int*)d_ws;  // B*N ints

    topk_rank_kernel<<<B_, SCAN_T, 0, stream>>>(idx, rank);

    const int nblocks = (B_ * N_) / ROWS_PER_BLK;  // 8192
    topk_scatter_kernel<<<nblocks, SCAT_T, 0, stream>>>(x, rank, out);
}